// DotInteraction_19456201851139
// MI455X (gfx1250) — compile-verified
//
#include <hip/hip_runtime.h>
#include <hip/hip_bf16.h>

typedef __attribute__((ext_vector_type(2))) float v2f;
typedef __attribute__((ext_vector_type(8))) float v8f;

#define BATCH   16384
#define NF      27
#define FD      128
#define NPAIR   351   // 27*26/2
#define WAVES_PER_BLOCK 8

// One wave32 computes one batch element's 27x27 Gram matrix as three 16x16
// WMMA tiles (C00 diag, C10 rect, C11 diag) using V_WMMA_F32_16X16X4_F32.
// For G = A*A^T the f32 A-fragment (16x4) and B-fragment (4x16, B = A^T)
// have identical per-lane register layouts, so one fragment feeds both
// operands: every input element is read from HBM exactly once.
__global__ __launch_bounds__(256)
void DotInteraction_19456201851139_kernel(const float* __restrict__ feat,
                                          float* __restrict__ out) {
    const int lane = threadIdx.x & 31;
    const int wave = threadIdx.x >> 5;
    const int b    = blockIdx.x * WAVES_PER_BLOCK + wave;

    const int half = lane >> 4;   // 0: K={0,1}, 1: K={2,3}
    const int l16  = lane & 15;

    // fragment row ownership
    const int r0 = l16;           // tile0 rows 0..15
    int r1 = 16 + l16;            // tile1 rows 16..31 (27..31 are padding)
    const float mask1 = (r1 < NF) ? 1.0f : 0.0f;
    if (r1 >= NF) r1 = NF - 1;    // clamp: duplicates lane 10's address (coalesces)

    const float* base = feat + (size_t)b * (NF * FD);
    const float* p0 = base + r0 * FD + 2 * half;
    const float* p1 = base + r1 * FD + 2 * half;

    v8f c00 = {};  // rows 0..15  x cols 0..15
    v8f c10 = {};  // rows 16..31 x cols 0..15
    v8f c11 = {};  // rows 16..31 x cols 16..31

    #pragma unroll 4
    for (int k0 = 0; k0 < FD; k0 += 4) {
        v2f a0 = *(const v2f*)(p0 + k0);
        v2f a1 = *(const v2f*)(p1 + k0);
        a1.x *= mask1;
        a1.y *= mask1;
        // 8 args: (neg_a, A, neg_b, B, c_mod, C, reuse_a, reuse_b)
        c00 = __builtin_amdgcn_wmma_f32_16x16x4_f32(false, a0, false, a0,
                                                    (short)0, c00, false, false);
        c10 = __builtin_amdgcn_wmma_f32_16x16x4_f32(false, a1, false, a0,
                                                    (short)0, c10, false, false);
        c11 = __builtin_amdgcn_wmma_f32_16x16x4_f32(false, a1, false, a1,
                                                    (short)0, c11, false, false);
    }

    // D layout (32-bit C/D 16x16): VGPR v of lane L holds M = v + 8*(L>>4),
    // N = L&15. Scatter the strict lower triangle: p = i*(i-1)/2 + j.
    float* ob = out + (size_t)b * NPAIR;
    const int j0 = l16;        // cols 0..15
    const int j1 = 16 + l16;   // cols 16..31
    #pragma unroll
    for (int v = 0; v < 8; ++v) {
        const int i0 = v + 8 * half;        // rows 0..15
        const int i1 = 16 + v + 8 * half;   // rows 16..31
        if (j0 < i0)
            ob[i0 * (i0 - 1) / 2 + j0] = c00[v];           // 120 elems
        if (i1 < NF)
            ob[i1 * (i1 - 1) / 2 + j0] = c10[v];           // 176 elems
        if (i1 < NF && j1 < i1)
            ob[i1 * (i1 - 1) / 2 + j1] = c11[v];           // 55 elems
    }
}

extern "C" void kernel_launch(void* const* d_in, const int* in_sizes, int n_in,
                              void* d_out, int out_size, void* d_ws, size_t ws_size,
                              hipStream_t stream) {
    (void)in_sizes; (void)n_in; (void)out_size; (void)d_ws; (void)ws_size;
    const float* feat = (const float*)d_in[0];
    float* out = (float*)d_out;
    dim3 grid(BATCH / WAVES_PER_BLOCK);
    dim3 block(WAVES_PER_BLOCK * 32);
    DotInteraction_19456201851139_kernel<<<grid, block, 0, stream>>>(feat, out);
}